// ConvLayer_6949257085117
// MI455X (gfx1250) — compile-verified
//
#include <hip/hip_runtime.h>

typedef float v2f __attribute__((ext_vector_type(2)));
typedef float v8f __attribute__((ext_vector_type(8)));

#define HID    128
#define N_USER 100000
#define N_BOOK 50000
#define N_EDGE 1000000

// ---------------------------------------------------------------------------
// Zero-fill workspace (graph-safe re-init every call).
// ---------------------------------------------------------------------------
__global__ void zero_kernel(float* __restrict__ p, long n) {
    long i      = (long)blockIdx.x * blockDim.x + threadIdx.x;
    long stride = (long)gridDim.x * blockDim.x;
    for (; i < n; i += stride) p[i] = 0.0f;
}

// ---------------------------------------------------------------------------
// Edge scatter: one wave32 per edge, one float4 (4 of 128 feats) per lane.
// Accumulates both directions (user->book and book->user) plus degree counts.
// ---------------------------------------------------------------------------
__global__ void __launch_bounds__(256) scatter_kernel(
    const float* __restrict__ x_user,
    const float* __restrict__ x_book,
    const int*   __restrict__ edge_src,
    const int*   __restrict__ edge_dst,
    float* __restrict__ sum_user,
    float* __restrict__ sum_book,
    float* __restrict__ cnt_user,
    float* __restrict__ cnt_book)
{
    const int wave = threadIdx.x >> 5;
    const int lane = threadIdx.x & 31;
    const int e    = blockIdx.x * 8 + wave;   // 8 waves per block, one edge each
    if (e >= N_EDGE) return;

    const int src = edge_src[e];  // uniform per wave (broadcast load)
    const int dst = edge_dst[e];

    const float4 u = ((const float4*)(x_user + (long)src * HID))[lane];
    const float4 b = ((const float4*)(x_book + (long)dst * HID))[lane];

    float* sb = sum_book + (long)dst * HID + lane * 4;
    float* su = sum_user + (long)src * HID + lane * 4;

    atomicAdd(sb + 0, u.x); atomicAdd(sb + 1, u.y);
    atomicAdd(sb + 2, u.z); atomicAdd(sb + 3, u.w);
    atomicAdd(su + 0, b.x); atomicAdd(su + 1, b.y);
    atomicAdd(su + 2, b.z); atomicAdd(su + 3, b.w);

    if (lane == 0) {
        atomicAdd(cnt_book + dst, 1.0f);
        atomicAdd(cnt_user + src, 1.0f);
    }
}

// ---------------------------------------------------------------------------
// out[m, :] = relu( (sum[m,:]/max(cnt[m],1)) @ Wl^T + bl + x[m,:] @ Wr^T )
//
// Block = 256 threads = 8 waves covering one 16-row stripe; wave w owns the
// 16x16 output tile at columns [16w, 16w+16). K=128 processed in 32 steps of
// V_WMMA_F32_16X16X4_F32, two WMMAs per step (Wl path + Wr path) into one
// shared f32 accumulator seeded with the bias.
//
// fp32 WMMA operand layouts (ISA 7.12.2):
//   A 16x4 : lane L (L<16): {A[M=L][k0], A[L][k0+1]}; lanes 16-31: K=k0+2,k0+3
//   B 4x16 : B[k][n] = W[n][k]  -> lane loads {W[n0+(L&15)][k0+2h], ...[+1]}
//   C/D    : VGPR v -> row m0 + v + 8*(L>>4), col n0 + (L&15)
// ---------------------------------------------------------------------------
__global__ void __launch_bounds__(256) sage_gemm_kernel(
    const float* __restrict__ sum,   // [M, HID] neighbor sums
    const float* __restrict__ cnt,   // [M] degree counts
    const float* __restrict__ x,     // [M, HID] root features
    const float* __restrict__ Wl,    // [HID, HID] row-major
    const float* __restrict__ bl,    // [HID]
    const float* __restrict__ Wr,    // [HID, HID] row-major
    float* __restrict__ out)         // [M, HID]
{
    const int wave = threadIdx.x >> 5;
    const int lane = threadIdx.x & 31;
    const int half = lane >> 4;      // 0: K pair {0,1}, 1: K pair {2,3}
    const int l15  = lane & 15;
    const int m0   = blockIdx.x * 16;
    const int n0   = wave * 16;

    const int rowA = m0 + l15;       // A-matrix row handled by this lane
    const int rowW = n0 + l15;       // weight row (= output column)

    const float inv = 1.0f / fmaxf(cnt[rowA], 1.0f);

    const float* aggp = sum + (long)rowA * HID + half * 2;
    const float* xp   = x   + (long)rowA * HID + half * 2;
    const float* wlp  = Wl  + (long)rowW * HID + half * 2;
    const float* wrp  = Wr  + (long)rowW * HID + half * 2;

    const float bias = bl[n0 + l15];
    v8f c;
#pragma unroll
    for (int v = 0; v < 8; ++v) c[v] = bias;

#pragma unroll 4
    for (int k = 0; k < HID; k += 4) {
        v2f a1 = *(const v2f*)(aggp + k);
        a1[0] *= inv;
        a1[1] *= inv;
        v2f a2 = *(const v2f*)(xp + k);
        v2f b1 = *(const v2f*)(wlp + k);
        v2f b2 = *(const v2f*)(wrp + k);
        // (neg_a, A, neg_b, B, c_mod, C, reuse_a, reuse_b)
        c = __builtin_amdgcn_wmma_f32_16x16x4_f32(false, a1, false, b1,
                                                  (short)0, c, false, false);
        c = __builtin_amdgcn_wmma_f32_16x16x4_f32(false, a2, false, b2,
                                                  (short)0, c, false, false);
    }

    float* op = out + (long)(m0 + half * 8) * HID + n0 + l15;
#pragma unroll
    for (int v = 0; v < 8; ++v) {
        op[(long)v * HID] = fmaxf(c[v], 0.0f);
    }
}

// ---------------------------------------------------------------------------
// Host-side launcher.
// Inputs (setup_inputs order): x_user, x_book, edge_src, edge_dst, Wl, bl, Wr
// Output: [out_user (100000x128) | out_book (50000x128)] fp32.
// ---------------------------------------------------------------------------
extern "C" void kernel_launch(void* const* d_in, const int* in_sizes, int n_in,
                              void* d_out, int out_size, void* d_ws, size_t ws_size,
                              hipStream_t stream) {
    const float* x_user   = (const float*)d_in[0];
    const float* x_book   = (const float*)d_in[1];
    const int*   edge_src = (const int*)  d_in[2];
    const int*   edge_dst = (const int*)  d_in[3];
    const float* Wl       = (const float*)d_in[4];
    const float* bl       = (const float*)d_in[5];
    const float* Wr       = (const float*)d_in[6];

    float* out_user = (float*)d_out;
    float* out_book = out_user + (size_t)N_USER * HID;

    float* ws       = (float*)d_ws;
    float* sum_user = ws;                                   // N_USER*HID
    float* sum_book = sum_user + (size_t)N_USER * HID;      // N_BOOK*HID
    float* cnt_user = sum_book + (size_t)N_BOOK * HID;      // N_USER
    float* cnt_book = cnt_user + N_USER;                    // N_BOOK

    const long total_ws = (long)N_USER * HID + (long)N_BOOK * HID + N_USER + N_BOOK;

    zero_kernel<<<4096, 256, 0, stream>>>(ws, total_ws);

    scatter_kernel<<<N_EDGE / 8, 256, 0, stream>>>(
        x_user, x_book, edge_src, edge_dst,
        sum_user, sum_book, cnt_user, cnt_book);

    sage_gemm_kernel<<<N_USER / 16, 256, 0, stream>>>(
        sum_user, cnt_user, x_user, Wl, bl, Wr, out_user);

    sage_gemm_kernel<<<N_BOOK / 16, 256, 0, stream>>>(
        sum_book, cnt_book, x_book, Wl, bl, Wr, out_book);
}